// AdditiveAttention_83013127897678
// MI455X (gfx1250) — compile-verified
//
#include <hip/hip_runtime.h>

// MI455X / gfx1250, wave32. Memory-bound (~537MB @ 23.3TB/s ~ 23us).
// GEMMs via V_WMMA_F32_16X16X4_F32; weights staged in LDS; B-fragments
// reused across 4 row-tiles per wave; 16 independent accumulator chains.
// tanh via hardware V_TANH_F32 when the builtin is available.

typedef __attribute__((ext_vector_type(2))) float v2f;
typedef __attribute__((ext_vector_type(8))) float v8f;

#if __has_builtin(__builtin_amdgcn_tanhf)
__device__ __forceinline__ float fast_tanh(float x) { return __builtin_amdgcn_tanhf(x); }
#else
__device__ __forceinline__ float fast_tanh(float x) { return tanhf(x); }
#endif

#define B_ 4
#define H_ 16
#define L_ 8192
#define D_ 64
#define ROWS (B_ * H_ * L_)      // 524288
#define TILES (ROWS / 16)        // 32768 (16-row WMMA tiles)
#define WAVES_PER_BLOCK 8
#define TILES_PER_WAVE 4
#define K1_BLOCKS (TILES / (WAVES_PER_BLOCK * TILES_PER_WAVE))  // 1024

// ---------------------------------------------------------------------------
// Kernel 1: per-row additive score e[row] = va . tanh(q@Wa^T + k@Ua^T + b) + vb,
// masked. Each wave computes 4 contiguous 16-row tiles (64 rows) with
// V_WMMA_F32_16X16X4_F32; B-fragments (weights) loaded once per kt from LDS.
// ---------------------------------------------------------------------------
__global__ __launch_bounds__(256) void score_kernel(
    const float* __restrict__ q, const float* __restrict__ k,
    const int* __restrict__ mask,
    const float* __restrict__ Wa_w, const float* __restrict__ Wa_b,
    const float* __restrict__ Ua_w, const float* __restrict__ Ua_b,
    const float* __restrict__ va_w, const float* __restrict__ va_b,
    float* __restrict__ e_out)
{
    __shared__ float sW[D_ * D_];   // Wa_w row-major [out][in]
    __shared__ float sU[D_ * D_];   // Ua_w
    __shared__ float sBsum[D_];     // Wa_b + Ua_b
    __shared__ float sVa[D_];       // va_w[0]

    const int tid = threadIdx.x;
    for (int i = tid; i < D_ * D_; i += 256) { sW[i] = Wa_w[i]; sU[i] = Ua_w[i]; }
    if (tid < D_) { sBsum[tid] = Wa_b[tid] + Ua_b[tid]; sVa[tid] = va_w[tid]; }
    __syncthreads();

    const float vb = va_b[0];
    const int wave = tid >> 5;
    const int lane = tid & 31;
    const int half = lane >> 4;     // K-pair select for A/B fragments
    const int nl   = lane & 15;     // M (A) or N (B/C) within tile

    const int tile0 = (blockIdx.x * WAVES_PER_BLOCK + wave) * TILES_PER_WAVE;
    const int row00 = tile0 * 16;   // first row of this wave's 64 contiguous rows

    // A-fragment base pointers; per-(t,kt) accesses are immediate offsets:
    // tile t -> +t*16*64 floats, kt -> +kt*4 floats (all within 24-bit IOFFSET).
    const float* aq = q + (size_t)(row00 + nl) * D_ + 2 * half;
    const float* ak = k + (size_t)(row00 + nl) * D_ + 2 * half;

    const v8f zero = {};
    v8f acc[TILES_PER_WAVE * 4];            // [tile][nt] -> 128 VGPRs
    #pragma unroll
    for (int i = 0; i < TILES_PER_WAVE * 4; ++i) acc[i] = zero;

    // Pass 1: q @ Wa^T   (B[k][n] = Wa_w[n][k], read transposed from LDS)
    #pragma unroll 4
    for (int kt = 0; kt < 16; ++kt) {
        v2f b[4];
        #pragma unroll
        for (int nt = 0; nt < 4; ++nt)
            b[nt] = *(const v2f*)(&sW[(nt * 16 + nl) * D_ + kt * 4 + 2 * half]);
        #pragma unroll
        for (int t = 0; t < TILES_PER_WAVE; ++t) {
            v2f a = *(const v2f*)(aq + t * (16 * D_) + kt * 4);
            #pragma unroll
            for (int nt = 0; nt < 4; ++nt)
                acc[t * 4 + nt] = __builtin_amdgcn_wmma_f32_16x16x4_f32(
                    false, a, false, b[nt], (short)0, acc[t * 4 + nt], false, false);
        }
    }
    // Pass 2: + k @ Ua^T into the same accumulators (we only need e1+e2)
    #pragma unroll 4
    for (int kt = 0; kt < 16; ++kt) {
        v2f b[4];
        #pragma unroll
        for (int nt = 0; nt < 4; ++nt)
            b[nt] = *(const v2f*)(&sU[(nt * 16 + nl) * D_ + kt * 4 + 2 * half]);
        #pragma unroll
        for (int t = 0; t < TILES_PER_WAVE; ++t) {
            v2f a = *(const v2f*)(ak + t * (16 * D_) + kt * 4);
            #pragma unroll
            for (int nt = 0; nt < 4; ++nt)
                acc[t * 4 + nt] = __builtin_amdgcn_wmma_f32_16x16x4_f32(
                    false, a, false, b[nt], (short)0, acc[t * 4 + nt], false, false);
        }
    }

    // Epilogue per tile: t = tanh(acc + bias); dot with va over the 64 dims.
    // C layout: VGPR r, lanes 0-15 -> M=r, N=nl; lanes 16-31 -> M=r+8, N=nl.
    #pragma unroll
    for (int t = 0; t < TILES_PER_WAVE; ++t) {
        float p[8];
        #pragma unroll
        for (int r = 0; r < 8; ++r) p[r] = 0.0f;
        #pragma unroll
        for (int nt = 0; nt < 4; ++nt) {
            const float bs = sBsum[nt * 16 + nl];
            const float va = sVa[nt * 16 + nl];
            #pragma unroll
            for (int r = 0; r < 8; ++r)
                p[r] += fast_tanh(acc[t * 4 + nt][r] + bs) * va;
        }
        #pragma unroll
        for (int r = 0; r < 8; ++r) {
            float s = p[r];
            s += __shfl_xor(s, 1);
            s += __shfl_xor(s, 2);
            s += __shfl_xor(s, 4);
            s += __shfl_xor(s, 8);
            if (nl == 0) {
                const int row = row00 + t * 16 + r + 8 * half;
                const float ev = s + vb;
                e_out[row] = (mask[row] == 0) ? -10000.0f : ev;
            }
        }
    }
}

// ---------------------------------------------------------------------------
// Kernel 2: per-(b,h) softmax statistics over L=8192 scores.
// ---------------------------------------------------------------------------
__global__ __launch_bounds__(256) void softmax_stats_kernel(
    const float* __restrict__ e, float* __restrict__ stats)
{
    __shared__ float red[256];
    const int row = blockIdx.x;                 // 0..63
    const int tid = threadIdx.x;
    const float* er = e + (size_t)row * L_;

    float m = -3.4e38f;
    for (int i = tid; i < L_; i += 256) m = fmaxf(m, er[i]);
    red[tid] = m; __syncthreads();
    for (int off = 128; off > 0; off >>= 1) {
        if (tid < off) red[tid] = fmaxf(red[tid], red[tid + off]);
        __syncthreads();
    }
    const float rmax = red[0];
    __syncthreads();

    float s = 0.0f;
    for (int i = tid; i < L_; i += 256) s += __expf(er[i] - rmax);
    red[tid] = s; __syncthreads();
    for (int off = 128; off > 0; off >>= 1) {
        if (tid < off) red[tid] += red[tid + off];
        __syncthreads();
    }
    if (tid == 0) { stats[row * 2] = rmax; stats[row * 2 + 1] = red[0]; }
}

// ---------------------------------------------------------------------------
// Kernel 3: out = (exp(e - max)/sum) * v, float4 streaming (bandwidth path).
// ---------------------------------------------------------------------------
__global__ __launch_bounds__(256) void scale_kernel(
    const float* __restrict__ v, const float* __restrict__ e,
    const float* __restrict__ stats, float* __restrict__ out)
{
    const size_t i4  = (size_t)blockIdx.x * 256 + threadIdx.x;  // float4 index
    const size_t row = i4 >> 4;            // 16 float4 per 64-elem row
    const int bh     = (int)(row >> 13);   // row / 8192
    const float m    = stats[bh * 2];
    const float s    = stats[bh * 2 + 1];
    const float alpha = __expf(e[row] - m) / s;

    float4 val = ((const float4*)v)[i4];
    val.x *= alpha; val.y *= alpha; val.z *= alpha; val.w *= alpha;
    ((float4*)out)[i4] = val;
}

// ---------------------------------------------------------------------------
extern "C" void kernel_launch(void* const* d_in, const int* in_sizes, int n_in,
                              void* d_out, int out_size, void* d_ws, size_t ws_size,
                              hipStream_t stream)
{
    const float* q    = (const float*)d_in[0];
    const float* k    = (const float*)d_in[1];
    const float* v    = (const float*)d_in[2];
    const int*   mask = (const int*)  d_in[3];
    const float* Wa_w = (const float*)d_in[4];
    const float* Wa_b = (const float*)d_in[5];
    const float* Ua_w = (const float*)d_in[6];
    const float* Ua_b = (const float*)d_in[7];
    const float* va_w = (const float*)d_in[8];
    const float* va_b = (const float*)d_in[9];

    float* e_ws  = (float*)d_ws;                                        // ROWS floats (2 MB)
    float* stats = (float*)((char*)d_ws + (size_t)ROWS * sizeof(float)); // 64*2 floats

    score_kernel<<<K1_BLOCKS, 256, 0, stream>>>(
        q, k, mask, Wa_w, Wa_b, Ua_w, Ua_b, va_w, va_b, e_ws);
    softmax_stats_kernel<<<B_ * H_, 256, 0, stream>>>(e_ws, stats);
    scale_kernel<<<(ROWS * (D_ / 4)) / 256, 256, 0, stream>>>(v, e_ws, stats, (float*)d_out);
}